// LocalGNN_36575941493470
// MI455X (gfx1250) — compile-verified
//
#include <hip/hip_runtime.h>
#include <hip/hip_bf16.h>

#define NN 4096
#define DIM 64
#define KCHUNK 64
#define A_STRIDE 68   // padded: A-frag b64 reads hit banks 4*row
#define U_STRIDE 68   // padded, keeps 16B alignment for b128 copies

#define AS1 __attribute__((address_space(1)))
#define AS3 __attribute__((address_space(3)))

typedef __attribute__((ext_vector_type(2))) float v2f;
typedef __attribute__((ext_vector_type(8))) float v8f;
typedef __attribute__((__vector_size__(4 * sizeof(int)))) int v4i;

__device__ __forceinline__ v8f wmma_f32_4(v2f a, v2f b, v8f c) {
    return __builtin_amdgcn_wmma_f32_16x16x4_f32(false, a, false, b, (short)0, c,
                                                 false, false);
}

// 16-byte async copy global -> LDS, tracked by ASYNCcnt (gfx1250 path)
__device__ __forceinline__ void async_copy_b128(const float* g, float* lds) {
#if __has_builtin(__builtin_amdgcn_global_load_async_to_lds_b128)
    __builtin_amdgcn_global_load_async_to_lds_b128((AS1 v4i*)g, (AS3 v4i*)lds,
                                                   0, 0);
#else
    asm volatile("global_load_async_to_lds_b128 %0, %1, off"
                 :: "v"((unsigned)(size_t)lds), "v"(g) : "memory");
#endif
}

__device__ __forceinline__ void wait_async0() {
#if __has_builtin(__builtin_amdgcn_s_wait_asynccnt)
    __builtin_amdgcn_s_wait_asynccnt(0);
#else
    asm volatile("s_wait_asynccnt 0x0" ::: "memory");
#endif
}

// ---------------------------------------------------------------------------
// Per row i: d_i = 1/(sqrt(1+rowsum)+1e-7); also u0[i][:] = d_i * x[i][:]
// one wave per row (butterfly leaves the full sum in every lane)
// ---------------------------------------------------------------------------
__global__ __launch_bounds__(256) void rowsum_scale_kernel(
        const float* __restrict__ graph, const float* __restrict__ x,
        float* __restrict__ dvec, float* __restrict__ u0) {
    const int row  = blockIdx.x * 8 + (threadIdx.x >> 5);  // [0, 4*4096)
    const int lane = threadIdx.x & 31;
    const float* g = graph + (size_t)row * NN;
    float s = 0.f;
    for (int j = lane * 4; j < NN; j += 32 * 4) {
        float4 v = *(const float4*)(g + j);
        s += v.x + v.y + v.z + v.w;
    }
    #pragma unroll
    for (int off = 16; off > 0; off >>= 1) s += __shfl_xor(s, off, 32);
    const float dval = 1.0f / (sqrtf(s + 1.0f) + 1e-7f);
    if (lane == 0) dvec[row] = dval;
    float2 xv = *(const float2*)(x + (size_t)row * DIM + lane * 2);
    *(float2*)(u0 + (size_t)row * DIM + lane * 2) =
        make_float2(xv.x * dval, xv.y * dval);
}

// ---------------------------------------------------------------------------
// acc_i = sum_j graph[i][j] * u_in[j]      (u_in already = d .* h_in)
// h_out[i] = d_i*acc_i + d_i*u_in[i]       (self-loop)
// u_out[i] = d_i*h_out[i]                  (feeds next pass)
// 128 threads / 4 waves; tile 16 x 64; async-double-buffered LDS
// ---------------------------------------------------------------------------
__device__ __forceinline__ void stage_chunk(const float* __restrict__ g,
                                            const float* __restrict__ u,
                                            int m0, int k0, int tid,
                                            float* __restrict__ abuf,
                                            float* __restrict__ ubuf) {
    const int ldr = tid >> 3;         // 0..15
    const int ldc = (tid & 7) * 4;    // 0..28
    const float* gr = g + (size_t)(m0 + ldr) * NN + k0 + ldc;
    async_copy_b128(gr,      abuf + ldr * A_STRIDE + ldc);
    async_copy_b128(gr + 32, abuf + ldr * A_STRIDE + ldc + 32);
    #pragma unroll
    for (int it = 0; it < 8; ++it) {
        int linear = it * 512 + tid * 4;     // 64x64 floats
        int kk = linear >> 6;
        int nn = linear & 63;
        async_copy_b128(u + (size_t)(k0 + kk) * DIM + nn,
                        ubuf + kk * U_STRIDE + nn);
    }
}

__global__ __launch_bounds__(128) void prop_kernel(const float* __restrict__ graph,
                                                   const float* __restrict__ u_in,
                                                   const float* __restrict__ dvec,
                                                   float* __restrict__ h_out,
                                                   float* __restrict__ u_out) {
    __shared__ float a_lds[2][16 * A_STRIDE];
    __shared__ float u_lds[2][KCHUNK * U_STRIDE];

    const int b    = blockIdx.y;
    const int m0   = blockIdx.x * 16;
    const int tid  = threadIdx.x;
    const int lane = tid & 31;
    const int wave = tid >> 5;
    const int n0   = wave * 16;

    const float* g  = graph + (size_t)b * NN * NN;
    const float* ui = u_in  + (size_t)b * NN * DIM;
    const float* d  = dvec  + (size_t)b * NN;
    float*       ho = h_out + (size_t)b * NN * DIM;
    float*       uo = u_out + (size_t)b * NN * DIM;

    const int fr = lane & 15;   // A: M / B,D: N
    const int hi = lane >> 4;   // K pair select & D row-half select

    v8f acc = {};

    stage_chunk(g, ui, m0, 0, tid, a_lds[0], u_lds[0]);
    wait_async0();
    __syncthreads();

    for (int k0 = 0; k0 < NN; k0 += KCHUNK) {
        const int cur = (k0 >> 6) & 1;
        const int nxt = cur ^ 1;
        if (k0 + KCHUNK < NN)
            stage_chunk(g, ui, m0, k0 + KCHUNK, tid, a_lds[nxt], u_lds[nxt]);

        const float* ab = a_lds[cur];
        const float* ub = u_lds[cur];
        #pragma unroll
        for (int kk = 0; kk < KCHUNK; kk += 4) {
            v2f a, bf;
            const float* ap = ab + fr * A_STRIDE + kk + 2 * hi;  // b64 read
            a.x = ap[0];
            a.y = ap[1];
            const float* bp = ub + (kk + 2 * hi) * U_STRIDE + n0 + fr;
            bf.x = bp[0];
            bf.y = bp[U_STRIDE];
            acc = wmma_f32_4(a, bf, acc);
        }
        wait_async0();
        __syncthreads();
    }

    const int n = n0 + fr;
    #pragma unroll
    for (int r = 0; r < 8; ++r) {
        int   m   = m0 + r + 8 * hi;
        float dm  = d[m];
        float uin = ui[(size_t)m * DIM + n];
        float hv  = dm * acc[r] + dm * uin;   // d_i*(G u)_i + d_i*u_i
        ho[(size_t)m * DIM + n] = hv;
        uo[(size_t)m * DIM + n] = dm * hv;
    }
}

// ---------------------------------------------------------------------------
// out[i][n] = bias[n] + sum_{c,k} h_c[i][k] * W[n][c*64+k]   (K = 256)
// ---------------------------------------------------------------------------
__global__ __launch_bounds__(128) void final_kernel(const float* __restrict__ h0,
                                                    const float* __restrict__ h1,
                                                    const float* __restrict__ h2,
                                                    const float* __restrict__ h3,
                                                    const float* __restrict__ W,
                                                    const float* __restrict__ bias,
                                                    float* __restrict__ out) {
    const int b    = blockIdx.y;
    const int m0   = blockIdx.x * 16;
    const int lane = threadIdx.x & 31;
    const int wave = threadIdx.x >> 5;
    const int n0   = wave * 16;
    const int fr   = lane & 15;
    const int hi   = lane >> 4;

    const size_t hoff = (size_t)b * NN * DIM;
    const float* hs[4] = {h0 + hoff, h1 + hoff, h2 + hoff, h3 + hoff};

    v8f acc = {};
    #pragma unroll
    for (int c = 0; c < 4; ++c) {
        const float* hc = hs[c];
        #pragma unroll
        for (int kk = 0; kk < 64; kk += 4) {
            v2f a, bf;
            const float* ap = hc + (size_t)(m0 + fr) * DIM + kk + 2 * hi;
            a.x = ap[0];
            a.y = ap[1];
            const float* wp = W + (size_t)(n0 + fr) * 256 + c * 64 + kk + 2 * hi;
            bf.x = wp[0];
            bf.y = wp[1];
            acc = wmma_f32_4(a, bf, acc);
        }
    }

    const int   n  = n0 + fr;
    const float bv = bias[n];
    float* o = out + (size_t)b * NN * DIM;
    #pragma unroll
    for (int r = 0; r < 8; ++r) {
        int m = m0 + r + 8 * hi;
        o[(size_t)m * DIM + n] = acc[r] + bv;
    }
}

extern "C" void kernel_launch(void* const* d_in, const int* in_sizes, int n_in,
                              void* d_out, int out_size, void* d_ws, size_t ws_size,
                              hipStream_t stream) {
    const float* x     = (const float*)d_in[0];   // [4,4096,64]
    const float* graph = (const float*)d_in[1];   // [4,4096,4096]
    const float* W     = (const float*)d_in[2];   // [64,256]
    const float* bias  = (const float*)d_in[3];   // [64]
    float* out = (float*)d_out;                   // [4,4096,64]

    const size_t HB = (size_t)4 * NN * DIM;       // 1M floats = 4MB
    char*  ws   = (char*)d_ws;
    float* dvec = (float*)ws;                     // 4*4096 floats = 64KB
    float* u0   = (float*)(ws + 64 * 1024);
    float* u1   = u0 + HB;
    float* u2   = u1 + HB;
    float* h1   = u2 + HB;
    float* h2   = h1 + HB;
    float* h3   = h2 + HB;

    rowsum_scale_kernel<<<4 * NN / 8, 256, 0, stream>>>(graph, x, dvec, u0);

    dim3 grid(NN / 16, 4);
    prop_kernel<<<grid, 128, 0, stream>>>(graph, u0, dvec, h1, u1);
    prop_kernel<<<grid, 128, 0, stream>>>(graph, u1, dvec, h2, u2);
    prop_kernel<<<grid, 128, 0, stream>>>(graph, u2, dvec, h3, u0); // u0 = scratch dump

    final_kernel<<<grid, 128, 0, stream>>>(x, h1, h2, h3, W, bias, out);
}